// PointToGridGNO_56839597195400
// MI455X (gfx1250) — compile-verified
//
#include <hip/hip_runtime.h>
#include <math.h>

// Problem sizes from setup_inputs(): B=2, M=32768, N=4096, D=128, K=8
#define B_ 2
#define M_ 32768
#define N_ 4096
#define D_ 128
#define K_ 8
#define HS 132  // padded LDS row stride (floats): (4r + c) mod 64 distinct over 16 rows

typedef __attribute__((ext_vector_type(2))) float v2f;
typedef __attribute__((ext_vector_type(8))) float v8f;

__device__ __forceinline__ float gelu_exact(float x) {
    return 0.5f * x * (1.0f + erff(x * 0.70710678118654752f));
}

// ---------------- Kernel 1: brute-force KNN (top-8 smallest distances) ----------------
__global__ __launch_bounds__(256) void knn_kernel(
    const float* __restrict__ grid, const float* __restrict__ pc,
    int* __restrict__ idx_out, float* __restrict__ w_out)
{
    __shared__ float sh_pc[N_ * 3];  // 48 KB
    const int b = blockIdx.y;
    const int tid = threadIdx.x;
    const float* pcb = pc + (size_t)b * N_ * 3;
    for (int i = tid; i < N_ * 3; i += 256) sh_pc[i] = pcb[i];
    __syncthreads();

    const int m = blockIdx.x * 256 + tid;
    const size_t gbase = ((size_t)b * M_ + m) * 3;
    const float gx = grid[gbase + 0];
    const float gy = grid[gbase + 1];
    const float gz = grid[gbase + 2];

    float bd[K_]; int bi[K_];
    #pragma unroll
    for (int j = 0; j < K_; ++j) { bd[j] = 3.4e38f; bi[j] = 0; }

    for (int n = 0; n < N_; ++n) {
        float dx = gx - sh_pc[3*n + 0];
        float dy = gy - sh_pc[3*n + 1];
        float dz = gz - sh_pc[3*n + 2];
        float d2 = fmaf(dx, dx, fmaf(dy, dy, dz*dz));
        if (d2 < bd[K_-1]) {                 // guard: rarely taken after warm-up
            bd[K_-1] = d2; bi[K_-1] = n;
            #pragma unroll
            for (int j = K_-1; j > 0; --j) { // fully unrolled bubble (static indices)
                float d0 = bd[j-1], d1 = bd[j];
                int   i0 = bi[j-1], i1 = bi[j];
                bool sw = d1 < d0;
                bd[j-1] = sw ? d1 : d0;  bd[j] = sw ? d0 : d1;
                bi[j-1] = sw ? i1 : i0;  bi[j] = sw ? i0 : i1;
            }
        }
    }

    float w[K_], s = 0.f;
    #pragma unroll
    for (int j = 0; j < K_; ++j) { w[j] = 1.0f / (sqrtf(bd[j]) + 1e-6f); s += w[j]; }
    const float inv = 1.0f / s;
    const size_t obase = ((size_t)b * M_ + m) * K_;
    #pragma unroll
    for (int j = 0; j < K_; ++j) { idx_out[obase + j] = bi[j]; w_out[obase + j] = w[j] * inv; }
}

// ---------------- Kernel 2: fused MLP (f32 WMMA) + K-reduction + LayerNorm ----------------
// 256 threads = 8 waves; wave w owns row-tile tr=w. A-operand (16 rows x 128 K)
// is hoisted into 32 v2f registers once per wave and reused across all 8 col-tiles.
__global__ __launch_bounds__(256) void gno_kernel(
    const float* __restrict__ grid, const float* __restrict__ pc,
    const float* __restrict__ feats,
    const float* __restrict__ W1, const float* __restrict__ b1,
    const float* __restrict__ W2, const float* __restrict__ b2,
    const float* __restrict__ gamma, const float* __restrict__ beta,
    const int* __restrict__ idx_ws, const float* __restrict__ w_ws,
    float* __restrict__ out)
{
    __shared__ float sh_h[128 * HS];     // h rows (g*8+k) x 128 cols, padded
    __shared__ float sh_w2t[128 * HS];   // W2 transposed: [n][k], padded
    __shared__ float sh_out[16 * 128];
    __shared__ float sh_W1[3 * 128];
    __shared__ float sh_b1[128], sh_b2[128], sh_gm[128], sh_bt[128];
    __shared__ int   sh_idx[128];
    __shared__ float sh_w[128];
    __shared__ float sh_mu[16], sh_rs[16];

    const int tid = threadIdx.x;   // 0..255
    const int b  = blockIdx.y;
    const int m0 = blockIdx.x * 16;

    // Stage weights (W2 transposed so B-operand k's are contiguous -> ds_load_b64)
    for (int i = tid; i < 128 * 128; i += 256) {
        int k = i >> 7, n = i & 127;
        sh_w2t[n * HS + k] = W2[i];
    }
    for (int i = tid; i < 3 * 128; i += 256) sh_W1[i] = W1[i];
    if (tid < 128) {
        sh_b1[tid] = b1[tid]; sh_b2[tid] = b2[tid];
        sh_gm[tid] = gamma[tid]; sh_bt[tid] = beta[tid];
        const size_t base = ((size_t)b * M_ + m0) * K_;  // tid = g*8+k
        sh_idx[tid] = idx_ws[base + tid];
        sh_w[tid]   = w_ws[base + tid];
    }
    __syncthreads();

    // h = GELU(rel_pos @ W1 + b1): thread tid owns row r=tid/2, column half tid%2
    {
        const int r = tid >> 1;
        const int g = r >> 3;
        const int pi = sh_idx[r];
        const size_t gb = ((size_t)b * M_ + m0 + g) * 3;
        const size_t pb = ((size_t)b * N_ + pi) * 3;
        const float rx = grid[gb + 0] - pc[pb + 0];
        const float ry = grid[gb + 1] - pc[pb + 1];
        const float rz = grid[gb + 2] - pc[pb + 2];
        float* hr = &sh_h[r * HS];
        const int c0 = (tid & 1) * 64;
        #pragma unroll 8
        for (int c = c0; c < c0 + 64; ++c) {
            float pre = fmaf(rx, sh_W1[c], fmaf(ry, sh_W1[128 + c], fmaf(rz, sh_W1[256 + c], sh_b1[c])));
            hr[c] = gelu_exact(pre);
        }
    }
    __syncthreads();

    // WMMA: wave owns row-tile tr=wave (rows 16*wave .. +15 => grid pts 2*wave, 2*wave+1)
    const int wave = tid >> 5, lane = tid & 31;
    const int lh = lane >> 4, lm = lane & 15;   // A/B: lane<16 -> K=v, lane>=16 -> K=v+2
    const int r0 = wave << 4;

    // Hoist full A-operand for this row-tile into registers (32 x v2f = 64 VGPRs)
    v2f areg[32];
    {
        const float* ha = &sh_h[(r0 + lm) * HS + 2 * lh];
        #pragma unroll
        for (int kk = 0; kk < 32; ++kk) areg[kk] = *(const v2f*)(ha + 4 * kk);
    }

    // Per-lane neighbor indices/weights for this lane's grid point (g = 2*wave + lh)
    const int gg = (wave << 1) + lh;
    unsigned int foff[8];
    float wv[8];
    #pragma unroll
    for (int j = 0; j < 8; ++j) {
        foff[j] = ((unsigned int)sh_idx[gg * 8 + j]) << 7;   // element offset pidx*D
        wv[j]   = sh_w[gg * 8 + j];
    }
    const float* featb = feats + (size_t)b * N_ * D_;        // uniform base -> SADDR form

    for (int tc = 0; tc < 8; ++tc) {
        const int c0 = tc << 4;
        const float* bb = &sh_w2t[(c0 + lm) * HS + 2 * lh];  // B[4kk + 2lh + v][n=lm]
        v8f acc = {};
        #pragma unroll
        for (int kk = 0; kk < 32; ++kk) {
            v2f bv = *(const v2f*)(bb + 4 * kk);
            acc = __builtin_amdgcn_wmma_f32_16x16x4_f32(
                false, areg[kk], false, bv, (short)0, acc, false, false);
        }
        // D layout: vgpr j -> row (j + 8*lh) => (g = 2*wave + lh, k = j), col = c0 + lm
        const int col = c0 + lm;
        const float b2v = sh_b2[col];
        float sum = 0.f;
        #pragma unroll
        for (int j = 0; j < 8; ++j) {
            float kap = acc[j] + b2v;
            sum = fmaf(kap * wv[j], featb[foff[j] + (unsigned int)col], sum);
        }
        sh_out[gg * 128 + col] = sum;
    }
    __syncthreads();

    // LayerNorm over D=128 for the 16 rows this WG owns
    if (tid < 16) {
        const float* r = &sh_out[tid * 128];
        float mu = 0.f;
        for (int c = 0; c < 128; ++c) mu += r[c];
        mu *= (1.0f / 128.0f);
        float var = 0.f;
        for (int c = 0; c < 128; ++c) { float dv = r[c] - mu; var = fmaf(dv, dv, var); }
        var *= (1.0f / 128.0f);
        sh_mu[tid] = mu;
        sh_rs[tid] = rsqrtf(var + 1e-5f);
    }
    __syncthreads();

    {
        const int row = tid >> 4, cs = (tid & 15) << 3;
        const float mu = sh_mu[row], rs = sh_rs[row];
        float* orow = out + ((size_t)b * M_ + m0 + row) * D_;
        #pragma unroll
        for (int c = cs; c < cs + 8; ++c) {
            orow[c] = (sh_out[row * 128 + c] - mu) * rs * sh_gm[c] + sh_bt[c];
        }
    }
}

extern "C" void kernel_launch(void* const* d_in, const int* in_sizes, int n_in,
                              void* d_out, int out_size, void* d_ws, size_t ws_size,
                              hipStream_t stream) {
    const float* grid  = (const float*)d_in[0];
    const float* pc    = (const float*)d_in[1];
    const float* feats = (const float*)d_in[2];
    const float* W1    = (const float*)d_in[3];
    const float* b1    = (const float*)d_in[4];
    const float* W2    = (const float*)d_in[5];
    const float* b2    = (const float*)d_in[6];
    const float* gamma = (const float*)d_in[7];
    const float* beta  = (const float*)d_in[8];
    float* out = (float*)d_out;

    int*   idx_ws = (int*)d_ws;                                // 2 MB
    float* w_ws   = (float*)(idx_ws + (size_t)B_ * M_ * K_);   // 2 MB

    dim3 g1(M_ / 256, B_);
    knn_kernel<<<g1, 256, 0, stream>>>(grid, pc, idx_ws, w_ws);

    dim3 g2(M_ / 16, B_);
    gno_kernel<<<g2, 256, 0, stream>>>(grid, pc, feats, W1, b1, W2, b2,
                                       gamma, beta, idx_ws, w_ws, out);
}